// BatchedChamferLoss_20486994002018
// MI455X (gfx1250) — compile-verified
//
#include <hip/hip_runtime.h>

typedef __attribute__((ext_vector_type(2))) float v2f;
typedef __attribute__((ext_vector_type(8))) float v8f;

#define BATCH 16
#define NPTS  4096
#define MPTS  4096
#define CHUNK 2048          // tgt points staged in LDS per pass (32 KB of float4)
#define WAVES 8             // 256 threads / wave32
#define TILES_PER_CHUNK (CHUNK / 16)
#define FLTMAX_BITS 0x7F7FFFFFu

// ---------------------------------------------------------------------------
// Kernel 0: init workspace + output (runs every launch; graph-safe)
// ---------------------------------------------------------------------------
__global__ void chamfer_init(unsigned* __restrict__ colmin,
                             float* __restrict__ rowsum,
                             float* __restrict__ out) {
    int i = blockIdx.x * blockDim.x + threadIdx.x;
    if (i < BATCH * MPTS) colmin[i] = FLTMAX_BITS;
    if (i < BATCH)        rowsum[i] = 0.0f;
    if (i == 0)           out[0]    = 0.0f;
}

// ---------------------------------------------------------------------------
// Kernel 1: fused distance tiles via V_WMMA_F32_16X16X4_F32 + min reductions
// One block = (batch b, 16-row strip of src) vs all 4096 tgt points.
// Per wave, one WMMA computes a full 16x16 d^2 tile:
//   A[n] = [-2sx,-2sy,-2sz, 1],  B[:,m] = [tx,ty,tz,|t|^2],  C[n][m] = |s|^2
// ---------------------------------------------------------------------------
__launch_bounds__(256, 2)
__global__ void chamfer_tiles(const float* __restrict__ src,
                              const float* __restrict__ tgt,
                              unsigned* __restrict__ colmin,
                              float* __restrict__ rowsum) {
    __shared__ float4   tl[CHUNK];        // (x, y, z, |t|^2)
    __shared__ unsigned rowmin_lds[16];

    const int blk   = blockIdx.x;
    const int b     = blk >> 8;           // / (NPTS/16)
    const int strip = blk & 255;
    const int n0    = strip << 4;

    const int tid  = threadIdx.x;
    const int lane = tid & 31;
    // wave index is uniform across the wave -- tell the compiler so the
    // m-tile loop gets scalar (s_cbranch) control and EXEC stays all-ones
    // at the WMMA (architectural requirement).
    const int wave = __builtin_amdgcn_readfirstlane(tid >> 5);
    const bool lo  = (lane < 16);
    const int r    = lane & 15;

    if (tid < 16) rowmin_lds[tid] = FLTMAX_BITS;

    // ---- Build A operand (16x4) and C (=|s|^2 broadcast per row) ----------
    // A lanes 0-15 hold K=0,1; lanes 16-31 hold K=2,3 (row = lane%16).
    const float* sb = src + ((size_t)b * NPTS + n0) * 3;
    const float sx = sb[r * 3 + 0];
    const float sy = sb[r * 3 + 1];
    const float sz = sb[r * 3 + 2];
    v2f a;
    a.x = lo ? (-2.0f * sx) : (-2.0f * sz);
    a.y = lo ? (-2.0f * sy) : 1.0f;
    const float s2 = sx * sx + sy * sy + sz * sz;  // lanes 0-15: s2[0..15]
    // C/D layout: VGPR i, lane l -> row i + 8*(l>=16), col l%16
    const int cbase = lo ? 0 : 8;
    v8f c;
#pragma unroll
    for (int i = 0; i < 8; ++i)
        c[i] = __shfl(s2, cbase + i, 32);

    // ---- Row-min accumulators (per lane: 8 rows x this lane's columns) ----
    float racc[8];
#pragma unroll
    for (int i = 0; i < 8; ++i) racc[i] = 3.4e38f;

    const float* tb = tgt + (size_t)b * MPTS * 3;
    unsigned* colbase = colmin + b * MPTS;
    const int ppoff = lo ? 0 : 2;   // lo lanes read (x,y); hi lanes read (z,t2)

    for (int ch = 0; ch < MPTS; ch += CHUNK) {
        __syncthreads();  // protect LDS from previous pass readers
        for (int i = tid; i < CHUNK; i += 256) {
            const int m = ch + i;
            const float x = tb[m * 3 + 0];
            const float y = tb[m * 3 + 1];
            const float z = tb[m * 3 + 2];
            tl[i] = make_float4(x, y, z, x * x + y * y + z * z);
        }
        __syncthreads();

        // 2 tiles per iteration: issue both B-operand LDS loads up front so
        // the second load's latency hides behind WMMA #1 + its min chain.
        for (int mt = wave; mt < TILES_PER_CHUNK; mt += 2 * WAVES) {
            const int m0 = mt << 4;
            const int m1 = m0 + (WAVES << 4);

            const float* p0 = (const float*)&tl[m0 + r] + ppoff;
            const float* p1 = (const float*)&tl[m1 + r] + ppoff;
            v2f bv0, bv1;
            bv0.x = p0[0]; bv0.y = p0[1];
            bv1.x = p1[0]; bv1.y = p1[1];

            v8f d0 = __builtin_amdgcn_wmma_f32_16x16x4_f32(
                false, a, false, bv0, (short)0, c, false, false);
            v8f d1 = __builtin_amdgcn_wmma_f32_16x16x4_f32(
                false, a, false, bv1, (short)0, c, false, false);

            float cmin0 = 3.4e38f, cmin1 = 3.4e38f;
#pragma unroll
            for (int i = 0; i < 8; ++i) {
                const float v0 = fmaxf(d0[i], 0.0f);   // numerical guard
                const float v1 = fmaxf(d1[i], 0.0f);
                racc[i] = fminf(racc[i], fminf(v0, v1));
                cmin0   = fminf(cmin0, v0);
                cmin1   = fminf(cmin1, v1);
            }
            // combine rows 0-7 (lanes 0-15) with rows 8-15 (lanes 16-31)
            cmin0 = fminf(cmin0, __shfl_xor(cmin0, 16, 32));
            cmin1 = fminf(cmin1, __shfl_xor(cmin1, 16, 32));
            if (lo) {
                atomicMin(&colbase[ch + m0 + lane], __float_as_uint(cmin0));
                atomicMin(&colbase[ch + m1 + lane], __float_as_uint(cmin1));
            }
        }
    }

    // ---- Finish row-min: reduce across the 16 columns held by each half ----
#pragma unroll
    for (int i = 0; i < 8; ++i) {
        float v = racc[i];
        v = fminf(v, __shfl_xor(v, 8, 32));
        v = fminf(v, __shfl_xor(v, 4, 32));
        v = fminf(v, __shfl_xor(v, 2, 32));
        v = fminf(v, __shfl_xor(v, 1, 32));
        if (lane == 0)  atomicMin(&rowmin_lds[i],     __float_as_uint(v));
        if (lane == 16) atomicMin(&rowmin_lds[i + 8], __float_as_uint(v));
    }
    __syncthreads();
    if (tid == 0) {
        float s = 0.0f;
#pragma unroll
        for (int i = 0; i < 16; ++i) s += __uint_as_float(rowmin_lds[i]);
        atomicAdd(&rowsum[b], s);
    }
}

// ---------------------------------------------------------------------------
// Kernel 2: per-batch means, weight, global mean
// ---------------------------------------------------------------------------
__global__ void chamfer_finalize(const unsigned* __restrict__ colmin,
                                 const float* __restrict__ rowsum,
                                 const float* __restrict__ weights,
                                 float* __restrict__ out) {
    __shared__ float part[WAVES];
    const int b   = blockIdx.x;
    const int tid = threadIdx.x;

    float s = 0.0f;
    for (int m = tid; m < MPTS; m += 256)
        s += __uint_as_float(colmin[b * MPTS + m]);
    for (int off = 16; off >= 1; off >>= 1)
        s += __shfl_xor(s, off, 32);
    if ((tid & 31) == 0) part[tid >> 5] = s;
    __syncthreads();
    if (tid == 0) {
        float tsum = 0.0f;
#pragma unroll
        for (int i = 0; i < WAVES; ++i) tsum += part[i];
        const float dist =
            (rowsum[b] * (1.0f / NPTS) + tsum * (1.0f / MPTS)) * weights[b];
        atomicAdd(out, dist * (1.0f / BATCH));
    }
}

// ---------------------------------------------------------------------------
extern "C" void kernel_launch(void* const* d_in, const int* in_sizes, int n_in,
                              void* d_out, int out_size, void* d_ws, size_t ws_size,
                              hipStream_t stream) {
    const float* src = (const float*)d_in[0];   // [B, N, 3]
    const float* tgt = (const float*)d_in[1];   // [B, M, 3]
    const float* w   = (const float*)d_in[2];   // [B]
    float* out = (float*)d_out;                 // scalar f32

    unsigned* colmin = (unsigned*)d_ws;                              // B*M u32
    float*    rowsum = (float*)((char*)d_ws +
                                (size_t)BATCH * MPTS * sizeof(unsigned)); // B f32

    const int initN = BATCH * MPTS;
    chamfer_init<<<(initN + 255) / 256, 256, 0, stream>>>(colmin, rowsum, out);

    chamfer_tiles<<<BATCH * (NPTS / 16), 256, 0, stream>>>(src, tgt, colmin,
                                                           rowsum);

    chamfer_finalize<<<BATCH, 256, 0, stream>>>(colmin, rowsum, w, out);
}